// CANLayer_49082886259395
// MI455X (gfx1250) — compile-verified
//
#include <hip/hip_runtime.h>

typedef float v2f __attribute__((ext_vector_type(2)));
typedef float v8f __attribute__((ext_vector_type(8)));

#define D 256

// =====================================================================
// GEMM: XM[N,256] = scale * (X[N,256] @ W[256,256])   (fp32 WMMA 16x16x4)
// One wave computes a 16x64 tile: 4 accumulators of 16x16.
// Requires N % 16 == 0 (true here: 50000 = 3125*16).
// =====================================================================
__global__ __launch_bounds__(256) void gemm_xw(const float* __restrict__ X,
                                               const float* __restrict__ W,
                                               float* __restrict__ XM,
                                               int nRowTiles, float scale)
{
    const int wave = (blockIdx.x * blockDim.x + threadIdx.x) >> 5;
    const int lane = threadIdx.x & 31;
    const int tm = wave >> 2;       // 16-row tile
    const int tn = wave & 3;        // 64-col tile
    if (tm >= nRowTiles) return;    // wave-uniform -> EXEC all ones inside

    const int m0   = tm << 4;
    const int n0   = tn << 6;
    const int mrow = lane & 15;
    const int kb   = (lane >> 4) << 1;   // 0 (lanes 0-15) or 2 (lanes 16-31)
    const int ncol = lane & 15;

#if __has_builtin(__builtin_amdgcn_wmma_f32_16x16x4_f32)
    v8f acc0 = {}, acc1 = {}, acc2 = {}, acc3 = {};
    const float* xr = X + (size_t)(m0 + mrow) * D + kb;
    for (int k0 = 0; k0 < D; k0 += 4) {
        v2f a;
        a.x = xr[k0];
        a.y = xr[k0 + 1];
        const float* wp = W + (size_t)(k0 + kb) * D + n0 + ncol;
        v2f b0, b1, b2, b3;
        b0.x = wp[0];  b0.y = wp[D];
        b1.x = wp[16]; b1.y = wp[D + 16];
        b2.x = wp[32]; b2.y = wp[D + 32];
        b3.x = wp[48]; b3.y = wp[D + 48];
        acc0 = __builtin_amdgcn_wmma_f32_16x16x4_f32(false, a, false, b0, (short)0, acc0, false, false);
        acc1 = __builtin_amdgcn_wmma_f32_16x16x4_f32(false, a, false, b1, (short)0, acc1, false, false);
        acc2 = __builtin_amdgcn_wmma_f32_16x16x4_f32(false, a, false, b2, (short)0, acc2, false, false);
        acc3 = __builtin_amdgcn_wmma_f32_16x16x4_f32(false, a, false, b3, (short)0, acc3, false, false);
    }
    // C/D layout: VGPR g holds row m0+g (lanes 0-15) / m0+g+8 (lanes 16-31), col = n0 + j*16 + (lane&15)
    const int rbase = m0 + ((lane >> 4) << 3);
    #pragma unroll
    for (int g = 0; g < 8; ++g) {
        float* orow = XM + (size_t)(rbase + g) * D + n0 + ncol;
        orow[0]  = acc0[g] * scale;
        orow[16] = acc1[g] * scale;
        orow[32] = acc2[g] * scale;
        orow[48] = acc3[g] * scale;
    }
#else
    // Scalar fallback (keeps kernel compiling if the fp32 WMMA builtin is absent;
    // histogram will show wmma==0 and we pivot to bf16/f16 WMMA next round).
    for (int rr = 0; rr < 16; ++rr) {
        int r = m0 + rr;
        for (int c = lane; c < 64; c += 32) {
            float acc = 0.f;
            for (int k = 0; k < D; ++k)
                acc += X[(size_t)r * D + k] * W[(size_t)k * D + n0 + c];
            XM[(size_t)r * D + n0 + c] = acc * scale;
        }
    }
#endif
}

// =====================================================================
// Per-node attention scores: st[i] = xm[i].attA, ss[i] = xm[i].attB
// One wave per row, float4 loads, shuffle reduction.
// =====================================================================
__global__ __launch_bounds__(256) void node_scores(const float* __restrict__ XM,
                                                   const float* __restrict__ att,
                                                   float* __restrict__ st,
                                                   float* __restrict__ ss, int N)
{
    const int wave = (blockIdx.x * blockDim.x + threadIdx.x) >> 5;
    const int lane = threadIdx.x & 31;
    if (wave >= N) return;
    const float4* row = (const float4*)(XM + (size_t)wave * D);
    const float4* aA  = (const float4*)att;
    const float4* aB  = (const float4*)(att + D);
    float s0 = 0.f, s1 = 0.f;
    for (int i = lane; i < D / 4; i += 32) {
        float4 v = row[i];
        float4 a = aA[i];
        float4 b = aB[i];
        s0 += v.x * a.x + v.y * a.y + v.z * a.z + v.w * a.w;
        s1 += v.x * b.x + v.y * b.y + v.z * b.z + v.w * b.w;
    }
    for (int off = 16; off > 0; off >>= 1) {
        s0 += __shfl_xor(s0, off, 32);
        s1 += __shfl_xor(s1, off, 32);
    }
    if (lane == 0) { st[wave] = s0; ss[wave] = s1; }
}

// =====================================================================
// Init per-target segment state for both convs (every call: graph-safe)
// =====================================================================
__global__ void init_seg(int* __restrict__ m0, float* __restrict__ s0,
                         int* __restrict__ m1, float* __restrict__ s1, int N)
{
    int i = blockIdx.x * blockDim.x + threadIdx.x;
    if (i < N) {
        m0[i] = (int)0x80000000; s0[i] = 0.f;
        m1[i] = (int)0x80000000; s1[i] = 0.f;
    }
}

__device__ __forceinline__ int float_flip(float f) {
    int i = __float_as_int(f);
    return i >= 0 ? i : (i ^ 0x7fffffff);   // order-preserving map (involution)
}
__device__ __forceinline__ float float_unflip(int i) {
    return __int_as_float(i >= 0 ? i : (i ^ 0x7fffffff));
}

// v[e] = elu(st[tgt] + ss[src]) * vals[e];  segmax[tgt] = max(...)
__global__ void edge_score_max(const int* __restrict__ tgt, const int* __restrict__ src,
                               const float* __restrict__ vals,
                               const float* __restrict__ st, const float* __restrict__ ss,
                               float* __restrict__ v, int* __restrict__ segmax, int E)
{
    int e = blockIdx.x * blockDim.x + threadIdx.x;
    if (e >= E) return;
    int t = tgt[e], s = src[e];
    float sc = st[t] + ss[s];
    sc = sc > 0.f ? sc : (__expf(sc) - 1.f);   // elu
    float ve = sc * vals[e];
    v[e] = ve;
    atomicMax(segmax + t, float_flip(ve));
}

// v[e] = exp(v[e] - segmax[tgt]);  segsum[tgt] += v[e]
__global__ void edge_exp_sum(const int* __restrict__ tgt,
                             float* __restrict__ v,
                             const int* __restrict__ segmax,
                             float* __restrict__ segsum, int E)
{
    int e = blockIdx.x * blockDim.x + threadIdx.x;
    if (e >= E) return;
    int t = tgt[e];
    float ex = __expf(v[e] - float_unflip(segmax[t]));
    v[e] = ex;
    atomicAdd(segsum + t, ex);
}

// OUT[tgt] += (v[e]/segsum[tgt]) * XM[src]   (one wave per edge, xm gathers hit L2)
__global__ __launch_bounds__(256) void edge_spmm(const int* __restrict__ tgt,
                                                 const int* __restrict__ src,
                                                 const float* __restrict__ v,
                                                 const float* __restrict__ segsum,
                                                 const float* __restrict__ XM,
                                                 float* __restrict__ OUT, int E)
{
    const int wave = (blockIdx.x * blockDim.x + threadIdx.x) >> 5;
    const int lane = threadIdx.x & 31;
    if (wave >= E) return;
    int t = tgt[wave], s = src[wave];
    float a = v[wave] / segsum[t];
    const float* xr = XM + (size_t)s * D;
    float* orow = OUT + (size_t)t * D;
    #pragma unroll
    for (int c = 0; c < 8; ++c)
        atomicAdd(orow + lane + c * 32, a * xr[lane + c * 32]);
}

__global__ void relu_k(float* __restrict__ o, int n4)
{
    int i = blockIdx.x * blockDim.x + threadIdx.x;
    if (i >= n4) return;
    float4* p = (float4*)o;
    float4 v = p[i];
    v.x = fmaxf(v.x, 0.f); v.y = fmaxf(v.y, 0.f);
    v.z = fmaxf(v.z, 0.f); v.w = fmaxf(v.w, 0.f);
    p[i] = v;
}

extern "C" void kernel_launch(void* const* d_in, const int* in_sizes, int n_in,
                              void* d_out, int out_size, void* d_ws, size_t ws_size,
                              hipStream_t stream)
{
    const float* x    = (const float*)d_in[0];
    const int*   li   = (const int*)d_in[1];   // [2,E]: row0 = tgt, row1 = src
    const float* lv   = (const float*)d_in[2];
    const int*   ui   = (const int*)d_in[3];
    const float* uv   = (const float*)d_in[4];
    const float* Wl   = (const float*)d_in[5];
    const float* al   = (const float*)d_in[6];
    const float* Wu   = (const float*)d_in[7];
    const float* au   = (const float*)d_in[8];
    const float* Wlin = (const float*)d_in[9];
    float* out = (float*)d_out;

    const int N = in_sizes[0] / D;
    const int E = in_sizes[2];
    const float EPS = 1.0f + 1e-06f;

    // workspace bump allocator (256B aligned)
    char* p = (char*)d_ws;
    auto alloc = [&](size_t bytes) { char* r = p; p += (bytes + 255) & ~(size_t)255; return r; };
    float* xm_l = (float*)alloc((size_t)N * D * sizeof(float));
    float* xm_u = (float*)alloc((size_t)N * D * sizeof(float));
    float* st_l = (float*)alloc((size_t)N * sizeof(float));
    float* ss_l = (float*)alloc((size_t)N * sizeof(float));
    float* st_u = (float*)alloc((size_t)N * sizeof(float));
    float* ss_u = (float*)alloc((size_t)N * sizeof(float));
    int*   sm_l = (int*)  alloc((size_t)N * sizeof(int));
    float* ssum_l = (float*)alloc((size_t)N * sizeof(float));
    int*   sm_u = (int*)  alloc((size_t)N * sizeof(int));
    float* ssum_u = (float*)alloc((size_t)N * sizeof(float));
    float* v_l  = (float*)alloc((size_t)E * sizeof(float));
    float* v_u  = (float*)alloc((size_t)E * sizeof(float));

    const int* tgt_l = li;     const int* src_l = li + E;
    const int* tgt_u = ui;     const int* src_u = ui + E;

    // --- dense phase: three WMMA GEMMs (W_lin fused with *EPS into d_out) ---
    const int rowTiles  = N / 16;                      // N % 16 == 0
    const int gemmWaves = rowTiles * 4;
    const int gemmBlk   = (gemmWaves + 7) / 8;         // 8 waves / 256-thread block
    gemm_xw<<<gemmBlk, 256, 0, stream>>>(x, Wl,   xm_l, rowTiles, 1.0f);
    gemm_xw<<<gemmBlk, 256, 0, stream>>>(x, Wu,   xm_u, rowTiles, 1.0f);
    gemm_xw<<<gemmBlk, 256, 0, stream>>>(x, Wlin, out,  rowTiles, EPS);

    // --- per-node attention scores ---
    const int scBlk = (N * 32 + 255) / 256;
    node_scores<<<scBlk, 256, 0, stream>>>(xm_l, al, st_l, ss_l, N);
    node_scores<<<scBlk, 256, 0, stream>>>(xm_u, au, st_u, ss_u, N);

    // --- segment state init (every call) ---
    init_seg<<<(N + 255) / 256, 256, 0, stream>>>(sm_l, ssum_l, sm_u, ssum_u, N);

    // --- sparse segment-softmax + SpMM, lower then upper, into d_out ---
    const int eBlk = (E + 255) / 256;
    const int eWaveBlk = (E * 32 + 255) / 256;
    edge_score_max<<<eBlk, 256, 0, stream>>>(tgt_l, src_l, lv, st_l, ss_l, v_l, sm_l, E);
    edge_exp_sum  <<<eBlk, 256, 0, stream>>>(tgt_l, v_l, sm_l, ssum_l, E);
    edge_spmm     <<<eWaveBlk, 256, 0, stream>>>(tgt_l, src_l, v_l, ssum_l, xm_l, out, E);

    edge_score_max<<<eBlk, 256, 0, stream>>>(tgt_u, src_u, uv, st_u, ss_u, v_u, sm_u, E);
    edge_exp_sum  <<<eBlk, 256, 0, stream>>>(tgt_u, v_u, sm_u, ssum_u, E);
    edge_spmm     <<<eWaveBlk, 256, 0, stream>>>(tgt_u, src_u, v_u, ssum_u, xm_u, out, E);

    // --- final ReLU ---
    const int n4 = (N * D) / 4;
    relu_k<<<(n4 + 255) / 256, 256, 0, stream>>>(out, n4);
}